// EpisodicMemory_43731357008356
// MI455X (gfx1250) — compile-verified
//
#include <hip/hip_runtime.h>
#include <math.h>
#include <stdint.h>

// Problem constants (match reference)
#define B_   4
#define L_   4096
#define DIM_ 256
#define S_   32
#define K_   8

typedef float v2f __attribute__((ext_vector_type(2)));
typedef float v8f __attribute__((ext_vector_type(8)));
typedef unsigned int u32;
typedef u32 v4u __attribute__((ext_vector_type(4)));
typedef int  v4i __attribute__((ext_vector_type(4)));
typedef int  v8i __attribute__((ext_vector_type(8)));

// ---------------------------------------------------------------------------
// Block-wide sum reduction (blockDim.x == 256)
// ---------------------------------------------------------------------------
__device__ __forceinline__ float blockReduceSum(float v, float* sm) {
    int tid = threadIdx.x;
    sm[tid] = v;
    __syncthreads();
    for (int off = 128; off > 0; off >>= 1) {
        if (tid < off) sm[tid] += sm[tid + off];
        __syncthreads();
    }
    float r = sm[0];
    __syncthreads();
    return r;
}

// ---------------------------------------------------------------------------
// K1: per-position avg complex magnitude + salience-head complex dot
// ---------------------------------------------------------------------------
__global__ void k_salience_pre(const float* __restrict__ z,
                               const float* __restrict__ salWr,
                               const float* __restrict__ salWi,
                               float* __restrict__ avg_mag,
                               float* __restrict__ phase_score) {
    __shared__ float sm[DIM_];
    int pos = blockIdx.x;          // b*L + l
    int d = threadIdx.x;
    size_t base = ((size_t)pos * DIM_ + d) * 2;
    float r = z[base], i = z[base + 1];
    float mag = sqrtf(r * r + i * i);
    float wr = salWr[d], wi = salWi[d];
    float pr = r * wr - i * wi;
    float pi = r * wi + i * wr;
    float msum = blockReduceSum(mag, sm);
    float prs  = blockReduceSum(pr, sm);
    float pis  = blockReduceSum(pi, sm);
    if (d == 0) {
        avg_mag[pos] = msum * (1.0f / DIM_);
        phase_score[pos] = sqrtf(prs * prs + pis * pis);
    }
}

// ---------------------------------------------------------------------------
// K2: avg_pool1d(k=5, count_include_pad) + sigmoid -> salience output
// ---------------------------------------------------------------------------
__global__ void k_salience(const float* __restrict__ avg_mag,
                           const float* __restrict__ phase_score,
                           const float* __restrict__ bias_p,
                           const float* __restrict__ nov_p,
                           float* __restrict__ salience) {
    int idx = blockIdx.x * blockDim.x + threadIdx.x;
    if (idx >= B_ * L_) return;
    int b = idx / L_, l = idx % L_;
    float lm = 0.f;
#pragma unroll
    for (int j = -2; j <= 2; ++j) {
        int ll = l + j;
        if (ll >= 0 && ll < L_) lm += avg_mag[b * L_ + ll];
    }
    lm *= 0.2f;
    float x = phase_score[idx] + (avg_mag[idx] - lm) * nov_p[0] + bias_p[0];
    salience[idx] = 1.0f / (1.0f + __expf(-x));
}

// ---------------------------------------------------------------------------
// K3: sequential span detection per batch; also emits event_valid + new_mask
// ---------------------------------------------------------------------------
__global__ void k_span_scan(const float* __restrict__ salience,
                            const float* __restrict__ slot_mask,
                            int* __restrict__ span_idx,
                            float* __restrict__ event_valid,
                            float* __restrict__ new_mask) {
    if (threadIdx.x != 0) return;
    int b = blockIdx.x;
    int cnt = 0;
    bool prev = false;
    for (int l = 0; l < L_; ++l) {
        bool above = salience[b * L_ + l] > 0.5f;
        if (above && !prev) cnt++;
        int sid = cnt - 1;
        span_idx[b * L_ + l] = (above && sid >= 0 && sid < S_) ? sid : -1;
        prev = above;
    }
    for (int s = 0; s < S_; ++s) {
        float ev = (s < cnt) ? 1.0f : 0.0f;
        event_valid[b * S_ + s] = ev;
        float nm = slot_mask[b * S_ + s] + ev;
        new_mask[b * S_ + s] = nm > 1.0f ? 1.0f : nm;
    }
}

// ---------------------------------------------------------------------------
// K4: salience-weighted segment pooling -> event_vecs [B,S,DIM,2]
// ---------------------------------------------------------------------------
__global__ void k_pool(const float* __restrict__ z,
                       const float* __restrict__ salience,
                       const int* __restrict__ span_idx,
                       float* __restrict__ event_vecs) {
    int b = blockIdx.x / S_;
    int s = blockIdx.x % S_;
    int d = threadIdx.x;
    float ar = 0.f, ai = 0.f, w = 0.f;
    for (int l = 0; l < L_; ++l) {
        if (span_idx[b * L_ + l] == s) {
            float sal = salience[b * L_ + l];
            size_t base = ((size_t)(b * L_ + l) * DIM_ + d) * 2;
            ar += sal * z[base];
            ai += sal * z[base + 1];
            w  += sal;
        }
    }
    float denom = fmaxf(w, 1e-8f);
    size_t ob = ((size_t)blockIdx.x * DIM_ + d) * 2;
    event_vecs[ob]     = ar / denom;
    event_vecs[ob + 1] = ai / denom;
}

// ---------------------------------------------------------------------------
// Complex GEMM 16x16 tile via V_WMMA_F32_16X16X4_F32 (global-A variant, used
// for the small event projections).
// ---------------------------------------------------------------------------
__device__ __forceinline__ void cgemm_tile_wmma(const float* __restrict__ X,
                                                const float* __restrict__ Wr,
                                                const float* __restrict__ Wi,
                                                int m0, int n0, int lane,
                                                v8f& out_r, v8f& out_i) {
    int row = lane & 15;
    int khalf = (lane >> 4) << 1;   // 0 for lanes 0-15, 2 for lanes 16-31
    v8f c_rr = {}; v8f c_ii = {}; v8f c_ri = {}; v8f c_ir = {};
    const float* xrow   = X  + ((size_t)(m0 + row) * DIM_) * 2;
    const float* wr_row = Wr +  (size_t)(n0 + row) * DIM_;
    const float* wi_row = Wi +  (size_t)(n0 + row) * DIM_;
    for (int kk = 0; kk < DIM_; kk += 4) {
        int k0 = kk + khalf;
        v2f a_r = { xrow[k0 * 2],     xrow[(k0 + 1) * 2] };
        v2f a_i = { xrow[k0 * 2 + 1], xrow[(k0 + 1) * 2 + 1] };
        v2f b_r = { wr_row[k0], wr_row[k0 + 1] };
        v2f b_i = { wi_row[k0], wi_row[k0 + 1] };
        c_rr = __builtin_amdgcn_wmma_f32_16x16x4_f32(false, a_r, false, b_r,
                                                     (short)0, c_rr, false, false);
        c_ii = __builtin_amdgcn_wmma_f32_16x16x4_f32(false, a_i, false, b_i,
                                                     (short)0, c_ii, false, false);
        c_ri = __builtin_amdgcn_wmma_f32_16x16x4_f32(false, a_r, false, b_i,
                                                     (short)0, c_ri, false, false);
        c_ir = __builtin_amdgcn_wmma_f32_16x16x4_f32(false, a_i, false, b_r,
                                                     (short)0, c_ir, false, false);
    }
    out_r = c_rr - c_ii;   // r@Wr.T - i@Wi.T
    out_i = c_ri + c_ir;   // r@Wi.T + i@Wr.T
}

// ---------------------------------------------------------------------------
// K5: event key/value projections (WMMA) + masked blend into new_keys/values
// ---------------------------------------------------------------------------
__global__ void k_event_proj(const float* __restrict__ event_vecs,
                             const float* __restrict__ ekWr, const float* __restrict__ ekWi,
                             const float* __restrict__ evWr, const float* __restrict__ evWi,
                             const float* __restrict__ slot_keys,
                             const float* __restrict__ slot_values,
                             const float* __restrict__ event_valid,
                             float* __restrict__ new_keys,
                             float* __restrict__ new_values) {
    int lane = threadIdx.x;
    int n0 = blockIdx.x * 16;
    int m0 = blockIdx.y * 16;
    v8f kr, ki, vr, vi;
    cgemm_tile_wmma(event_vecs, ekWr, ekWi, m0, n0, lane, kr, ki);
    cgemm_tile_wmma(event_vecs, evWr, evWi, m0, n0, lane, vr, vi);
    int nlane = lane & 15;
    int mofs = (lane >> 4) << 3;
#pragma unroll
    for (int v = 0; v < 8; ++v) {
        int m = m0 + mofs + v;      // m = b*S + s
        int n = n0 + nlane;
        float val = event_valid[m];
        size_t o = ((size_t)m * DIM_ + n) * 2;
        new_keys[o]       = val * kr[v] + (1.0f - val) * slot_keys[o];
        new_keys[o + 1]   = val * ki[v] + (1.0f - val) * slot_keys[o + 1];
        new_values[o]     = val * vr[v] + (1.0f - val) * slot_values[o];
        new_values[o + 1] = val * vi[v] + (1.0f - val) * slot_values[o + 1];
    }
}

// ---------------------------------------------------------------------------
// K6: key magnitudes
// ---------------------------------------------------------------------------
__global__ void k_kmag(const float* __restrict__ new_keys, float* __restrict__ kmag) {
    __shared__ float sm[DIM_];
    int bs = blockIdx.x;
    int d = threadIdx.x;
    size_t o = ((size_t)bs * DIM_ + d) * 2;
    float r = new_keys[o], i = new_keys[o + 1];
    float s = blockReduceSum(r * r + i * i, sm);
    if (d == 0) kmag[bs] = sqrtf(s + 1e-8f);
}

// ---------------------------------------------------------------------------
// K7: big query complex GEMM [B*L,256]x[256,256].
// Block = 128 threads (4 waves) owns one 16-row M tile.
//  * The 16x512-float A tile is DMA'd into LDS ONCE via the Tensor Data Mover
//    (2D descriptor, 1-DWORD pad per 256 DWORDs to spread LDS banks), waited
//    with s_wait_tensorcnt. This cuts A global traffic 16x vs the naive form
//    (0.54 GB -> 33.5 MB, i.e. to the HBM roofline floor at 23.3 TB/s).
//  * Each wave then computes 4 N-tiles from LDS-resident A; W rows stream
//    from L2 (W working set is 512 KB << 192 MB L2).
// LDS float index of element (row, k, c): row*514 + 2k + c + (k>>7)
// ---------------------------------------------------------------------------
#define LDSROW 514
__global__ void __launch_bounds__(128) k_query_gemm(const float* __restrict__ z,
                                                    const float* __restrict__ Wr,
                                                    const float* __restrict__ Wi,
                                                    float* __restrict__ q) {
    __shared__ float lds_a[16 * LDSROW];
    int tid  = threadIdx.x;
    int wave = tid >> 5;
    int lane = tid & 31;
    int m0 = blockIdx.x * 16;

    // ---- TDM path: one DMA per block, issued by wave 0 ----
    // (this toolchain exposes the 6-arg tensor_load_to_lds: g0,g1,g2,g3,extra,cpol)
    if (wave == 0) {
        uint64_t ga = (uint64_t)(uintptr_t)(const void*)(z + (size_t)m0 * (DIM_ * 2));
        u32 lds_off = (u32)(uintptr_t)(void*)&lds_a[0];
        // D# group 0: count=1 | lds_addr | global_addr | type=2
        v4u g0;
        g0[0] = 1u;
        g0[1] = lds_off;
        g0[2] = (u32)(ga & 0xffffffffu);
        g0[3] = (u32)((ga >> 32) & 0x01ffffffu) | 0x80000000u;   // type=2 in [127:126]
        // D# group 1: data_size=4B, pad_enable, pad_interval=256dw, pad_amount=1dw,
        //             tensor_dim0=512, tensor_dim1=16, tile_dim0=512, tile_dim1=16,
        //             tensor_dim0_stride=512
        v8i g1;
        g1[0] = (int)((2u << 16) | (1u << 20) | (7u << 22) | (0u << 25));
        g1[1] = (int)((512u & 0xffffu) << 16);          // tensor_dim0[15:0]
        g1[2] = (int)((512u >> 16) | (16u << 16));      // tensor_dim0[31:16] | tensor_dim1[15:0]
        g1[3] = (int)((512u & 0xffffu) << 16);          // tensor_dim1[31:16]=0 | tile_dim0
        g1[4] = (int)(16u);                             // tile_dim1=16, tile_dim2=0
        g1[5] = (int)(512u);                            // tensor_dim0_stride[31:0]
        g1[6] = 0;
        g1[7] = 0;
        v4i gz4 = {0, 0, 0, 0};
        v8i gz8 = {0, 0, 0, 0, 0, 0, 0, 0};
        __builtin_amdgcn_tensor_load_to_lds(g0, g1, gz4, gz4, gz8, 0);
        __builtin_amdgcn_s_wait_tensorcnt(0);
    }
    __syncthreads();

    int row = lane & 15;
    int khalf = (lane >> 4) << 1;
    const float* arow = &lds_a[row * LDSROW];

    for (int jn = 0; jn < 4; ++jn) {
        int n0 = (wave * 4 + jn) * 16;
        const float* wr_row = Wr + (size_t)(n0 + row) * DIM_;
        const float* wi_row = Wi + (size_t)(n0 + row) * DIM_;
        __builtin_prefetch(wr_row, 0, 0);
        __builtin_prefetch(wi_row, 0, 0);
        v8f c_rr = {}; v8f c_ii = {}; v8f c_ri = {}; v8f c_ir = {};
        for (int kk = 0; kk < DIM_; kk += 4) {
            int k0 = kk + khalf;
            int i0 = k0 * 2 + (k0 >> 7);
            int i1 = (k0 + 1) * 2 + ((k0 + 1) >> 7);
            v2f a_r = { arow[i0],     arow[i1] };
            v2f a_i = { arow[i0 + 1], arow[i1 + 1] };
            v2f b_r = { wr_row[k0], wr_row[k0 + 1] };
            v2f b_i = { wi_row[k0], wi_row[k0 + 1] };
            c_rr = __builtin_amdgcn_wmma_f32_16x16x4_f32(false, a_r, false, b_r,
                                                         (short)0, c_rr, false, false);
            c_ii = __builtin_amdgcn_wmma_f32_16x16x4_f32(false, a_i, false, b_i,
                                                         (short)0, c_ii, false, false);
            c_ri = __builtin_amdgcn_wmma_f32_16x16x4_f32(false, a_r, false, b_i,
                                                         (short)0, c_ri, false, false);
            c_ir = __builtin_amdgcn_wmma_f32_16x16x4_f32(false, a_i, false, b_r,
                                                         (short)0, c_ir, false, false);
        }
        v8f qr = c_rr - c_ii;
        v8f qi = c_ri + c_ir;
        int nlane = lane & 15;
        int mofs = (lane >> 4) << 3;
#pragma unroll
        for (int v = 0; v < 8; ++v) {
            int m = m0 + mofs + v;
            int n = n0 + nlane;
            size_t o = ((size_t)m * DIM_ + n) * 2;
            q[o]     = qr[v];
            q[o + 1] = qi[v];
        }
    }
}

// ---------------------------------------------------------------------------
// K8: cosine scores vs 32 slots, top-8, softmax, retrieval, RMS norm -> out
// ---------------------------------------------------------------------------
__global__ void k_attend(const float* __restrict__ q,
                         const float* __restrict__ new_keys,
                         const float* __restrict__ new_values,
                         const float* __restrict__ new_mask,
                         const float* __restrict__ kmag,
                         const float* __restrict__ gamma,
                         float* __restrict__ out) {
    __shared__ float sm[DIM_];
    __shared__ float s_scores[S_];
    __shared__ int   s_idx[K_];
    __shared__ float s_attn[K_];
    int pos = blockIdx.x;            // b*L + l
    int b = pos / L_;
    int tid = threadIdx.x;
    size_t qb = (size_t)pos * DIM_ * 2;

    float qr_d = q[qb + tid * 2], qi_d = q[qb + tid * 2 + 1];
    float qsum = blockReduceSum(qr_d * qr_d + qi_d * qi_d, sm);
    float qmag = sqrtf(qsum + 1e-8f);

    int s = tid >> 3;
    int sub = tid & 7;
    const float* krow = new_keys + ((size_t)(b * S_ + s) * DIM_) * 2;
    float dot = 0.f;
#pragma unroll 4
    for (int j = 0; j < DIM_ / 8; ++j) {
        int d = sub * (DIM_ / 8) + j;
        dot += q[qb + d * 2] * krow[d * 2] + q[qb + d * 2 + 1] * krow[d * 2 + 1];
    }
    for (int off = 4; off > 0; off >>= 1)
        dot += __shfl_down(dot, off, 8);
    if (sub == 0) {
        float score = dot / (qmag * kmag[b * S_ + s] + 1e-8f);
        if (new_mask[b * S_ + s] == 0.0f) score = -1e9f;
        s_scores[s] = score;
    }
    __syncthreads();

    if (tid == 0) {
        float local[S_];
        for (int j = 0; j < S_; ++j) local[j] = s_scores[j];
        float tv[K_]; int ti[K_];
        for (int k = 0; k < K_; ++k) {
            int best = 0; float bv = -3e38f;
            for (int j = 0; j < S_; ++j)
                if (local[j] > bv) { bv = local[j]; best = j; }
            tv[k] = bv; ti[k] = best; local[best] = -3e38f;
        }
        float mx = tv[0];
        float sum = 0.f;
        for (int k = 0; k < K_; ++k) { tv[k] = __expf(tv[k] - mx); sum += tv[k]; }
        float inv = 1.0f / sum;
        for (int k = 0; k < K_; ++k) { s_attn[k] = tv[k] * inv; s_idx[k] = ti[k]; }
    }
    __syncthreads();

    float rr = 0.f, ri = 0.f;
#pragma unroll
    for (int k = 0; k < K_; ++k) {
        const float* vrow = new_values + ((size_t)(b * S_ + s_idx[k]) * DIM_) * 2;
        float a = s_attn[k];
        rr += a * vrow[tid * 2];
        ri += a * vrow[tid * 2 + 1];
    }

    float m2 = blockReduceSum(rr * rr + ri * ri, sm) * (1.0f / DIM_);
    float scale = gamma[tid] * rsqrtf(m2 + 1e-8f);
    out[qb + tid * 2]     = rr * scale;
    out[qb + tid * 2 + 1] = ri * scale;
}

// ---------------------------------------------------------------------------
// Launch
// ---------------------------------------------------------------------------
extern "C" void kernel_launch(void* const* d_in, const int* in_sizes, int n_in,
                              void* d_out, int out_size, void* d_ws, size_t ws_size,
                              hipStream_t stream) {
    const float* z             = (const float*)d_in[0];
    const float* slot_keys     = (const float*)d_in[1];
    const float* slot_values   = (const float*)d_in[2];
    const float* slot_mask     = (const float*)d_in[3];
    const float* sal_Wr        = (const float*)d_in[4];
    const float* sal_Wi        = (const float*)d_in[5];
    const float* score_bias    = (const float*)d_in[6];
    const float* novelty_scale = (const float*)d_in[7];
    const float* ek_Wr         = (const float*)d_in[8];
    const float* ek_Wi         = (const float*)d_in[9];
    const float* ev_Wr         = (const float*)d_in[10];
    const float* ev_Wi         = (const float*)d_in[11];
    const float* rq_Wr         = (const float*)d_in[12];
    const float* rq_Wi         = (const float*)d_in[13];
    const float* gamma         = (const float*)d_in[14];

    // Output regions, concatenated flat in return order
    float* out    = (float*)d_out;
    float* o_out  = out;                                   // [B,L,DIM,2]
    float* o_keys = o_out  + (size_t)B_ * L_ * DIM_ * 2;   // [B,S,DIM,2]
    float* o_vals = o_keys + (size_t)B_ * S_ * DIM_ * 2;   // [B,S,DIM,2]
    float* o_mask = o_vals + (size_t)B_ * S_ * DIM_ * 2;   // [B,S]
    float* o_sal  = o_mask + B_ * S_;                      // [B,L]

    // Workspace layout
    float* ws       = (float*)d_ws;
    float* w_avg    = ws;                                   // B*L
    float* w_phase  = w_avg + B_ * L_;                      // B*L
    int*   w_span   = (int*)(w_phase + B_ * L_);            // B*L ints
    float* w_evalid = (float*)(w_span + B_ * L_);           // B*S
    float* w_evecs  = w_evalid + B_ * S_;                   // B*S*DIM*2
    float* w_kmag   = w_evecs + (size_t)B_ * S_ * DIM_ * 2; // B*S
    float* w_q      = w_kmag + B_ * S_;                     // B*L*DIM*2

    k_salience_pre<<<B_ * L_, DIM_, 0, stream>>>(z, sal_Wr, sal_Wi, w_avg, w_phase);
    k_salience<<<(B_ * L_ + 255) / 256, 256, 0, stream>>>(w_avg, w_phase, score_bias,
                                                          novelty_scale, o_sal);
    k_span_scan<<<B_, 32, 0, stream>>>(o_sal, slot_mask, w_span, w_evalid, o_mask);
    k_pool<<<B_ * S_, DIM_, 0, stream>>>(z, o_sal, w_span, w_evecs);
    k_event_proj<<<dim3(DIM_ / 16, (B_ * S_) / 16), 32, 0, stream>>>(
        w_evecs, ek_Wr, ek_Wi, ev_Wr, ev_Wi, slot_keys, slot_values, w_evalid,
        o_keys, o_vals);
    k_kmag<<<B_ * S_, DIM_, 0, stream>>>(o_keys, w_kmag);
    k_query_gemm<<<(B_ * L_) / 16, 128, 0, stream>>>(z, rq_Wr, rq_Wi, w_q);
    k_attend<<<B_ * L_, DIM_, 0, stream>>>(w_q, o_keys, o_vals, o_mask, w_kmag,
                                           gamma, o_out);
}